// SinkhornLayer_27857157882053
// MI455X (gfx1250) — compile-verified
//
#include <hip/hip_runtime.h>
#include <hip/hip_bf16.h>
#include <stdint.h>

// ---------------------------------------------------------------------------
// Sinkhorn (log-domain) via dual potentials on MI455X / gfx1250.
//   la0 = -C/eps stays implicit; only u (row) and v (col) potentials evolve.
//   row pass: u_i = log sum_j exp(-C_ij/eps - v_j)      (WMMA f32 row sums)
//   col pass: v_j = log sum_i exp(-C_ij/eps - u_i)      (thread-per-column)
//   output  : exp(-C/eps - u_i - v_j)
// Roofline: dual-potential form reads C 41x (mostly L2: 4MB/slice, 192MB L2)
// and writes 256MB once -> ~0.5GB HBM; bottleneck is ~2.6G v_exp_f32 ops,
// so the LSE summation is offloaded to the matrix pipe (wmma f32 16x16x4).
// ---------------------------------------------------------------------------

typedef float v2f __attribute__((ext_vector_type(2)));
typedef float v8f __attribute__((ext_vector_type(8)));
typedef unsigned int v4u __attribute__((ext_vector_type(4)));
typedef int v4i __attribute__((ext_vector_type(4)));
typedef int v8i __attribute__((ext_vector_type(8)));

#define NB   64      // batches
#define NN   1024    // matrix dim
#define PITCH 1028   // LDS row pitch in dwords: 1024 + 4 pad dwords (1 per 256)

// ----------------------------------------------------------------- init v = 0
__global__ __launch_bounds__(256) void sink_init(float* __restrict__ v) {
  v[blockIdx.x * 256 + threadIdx.x] = 0.0f;
}

// ----------------------------------------------------------------- row pass
// Block: 256 threads (8 waves). Handles 16 rows x 1024 cols of batch b.
// TDM stages the tile into LDS with 1-dword padding every 256 dwords so the
// WMMA A-fragment ds_load_b64 reads (bank = 4*m + 2*hi + const) are
// conflict-free. Wave w owns columns [w*128, w*128+128).
__global__ __launch_bounds__(256) void sink_row(const float* __restrict__ C,
                                                const float* __restrict__ eps,
                                                const float* __restrict__ v,
                                                float* __restrict__ u) {
  __shared__ float sv[NN];                          // v values
  __shared__ __align__(16) float tile[16 * PITCH];  // padded C tile (TDM dest)
  __shared__ float part[8 * 16];                    // per-wave row partials

  const int b  = blockIdx.y;
  const int r0 = blockIdx.x * 16;
  const float inv_eps = 1.0f / eps[0];

  // stage v for this batch
  for (int j = threadIdx.x; j < NN; j += 256) sv[j] = v[b * NN + j];

  // --- TDM: DMA the 16x1024 f32 tile into LDS (one wave issues) -----------
  if (threadIdx.x < 32) {
    const unsigned lds_addr = (unsigned)(uintptr_t)&tile[0]; // low 32b = LDS off
    const unsigned long long ga =
        (unsigned long long)(uintptr_t)(C + ((size_t)b * NN + r0) * NN);
    v4u g0;
    g0[0] = 1u;                                   // count=1, user descriptor
    g0[1] = lds_addr;                             // lds_addr (bytes)
    g0[2] = (unsigned)(ga & 0xFFFFFFFFu);         // global_addr[31:0]
    g0[3] = (unsigned)((ga >> 32) & 0x1FFFFFFu)   // global_addr[56:32]
            | (2u << 30);                         // type = 2 ("image")
    v8i g1;
    g1[0] = (2 << 16)        // data_size = 4B
          | (1 << 20)        // pad_enable
          | (7 << 22)        // pad_interval: every 256 dwords
          | (0 << 25);       // pad_amount: 1 dword  -> pitch 1028 dwords/row
    g1[1] = (NN & 0xFFFF) << 16;                  // tensor_dim0[15:0] @bit48
    g1[2] = ((NN >> 16) & 0xFFFF) | ((NN & 0xFFFF) << 16); // dim0 hi | dim1 lo
    g1[3] = ((NN >> 16) & 0xFFFF) | (NN << 16);   // dim1 hi | tile_dim0 = 1024
    g1[4] = 16;                                   // tile_dim1 = 16, tile_dim2 = 0
    g1[5] = NN;                                   // tensor_dim0_stride lo32
    g1[6] = 0;
    g1[7] = 0;
    v4i g2 = {0, 0, 0, 0};                        // 2D tensor: groups 2/3 unused
    v4i g3 = {0, 0, 0, 0};
    v8i g4 = {0, 0, 0, 0, 0, 0, 0, 0};            // extra group (clang-23 form)
    __builtin_amdgcn_tensor_load_to_lds(g0, g1, g2, g3, g4, 0);
    __builtin_amdgcn_s_wait_tensorcnt(0);
  }
  __syncthreads();

  // --- exp + WMMA row-sum accumulation ------------------------------------
  const int lane = threadIdx.x & 31;
  const int wave = threadIdx.x >> 5;
  const int m    = lane & 15;    // A-matrix row (M)
  const int hi   = lane >> 4;    // lanes 16-31 hold K = 2,3
  const int c0   = wave * 128;   // column range of this wave

  const float* trow = tile + m * PITCH + (c0 >> 8); // +intra-row pad offset
  const v2f ones = {1.0f, 1.0f};
  v8f acc0 = {};                 // two accumulators -> two overlapped XDL chains
  v8f acc1 = {};
#pragma unroll 4
  for (int jt = 0; jt < 32; jt += 2) {
    const int j0 = c0 + jt * 4 + 2 * hi;          // K = 2*hi + {0,1}
    const int j1 = j0 + 4;
    v2f cA = *(const v2f*)(trow + j0);            // ds_load_b64, conflict-free
    v2f vA = *(const v2f*)(sv + j0);              // broadcast read
    v2f cB = *(const v2f*)(trow + j1);
    v2f vB = *(const v2f*)(sv + j1);
    v2f a0, a1;
    a0.x = __expf(fmaf(cA.x, -inv_eps, -vA.x));
    a0.y = __expf(fmaf(cA.y, -inv_eps, -vA.y));
    a1.x = __expf(fmaf(cB.x, -inv_eps, -vB.x));
    a1.y = __expf(fmaf(cB.y, -inv_eps, -vB.y));
    // D[m][n] = sum_k A[m][k]*1 + C[m][n]  -> every column n = row sum
    acc0 = __builtin_amdgcn_wmma_f32_16x16x4_f32(
        false, a0, false, ones, (short)0, acc0, false, false);
    acc1 = __builtin_amdgcn_wmma_f32_16x16x4_f32(
        false, a1, false, ones, (short)0, acc1, false, false);
  }
  const v8f acc = acc0 + acc1;

  // D layout: vgpr r = row r (lanes 0-15) / row 8+r (lanes 16-31)
  if (m == 0) {
    const int rbase = wave * 16 + hi * 8;
    part[rbase + 0] = acc[0]; part[rbase + 1] = acc[1];
    part[rbase + 2] = acc[2]; part[rbase + 3] = acc[3];
    part[rbase + 4] = acc[4]; part[rbase + 5] = acc[5];
    part[rbase + 6] = acc[6]; part[rbase + 7] = acc[7];
  }
  __syncthreads();

  if (threadIdx.x < 16) {
    float s = 0.0f;
#pragma unroll
    for (int w = 0; w < 8; ++w) s += part[w * 16 + threadIdx.x];
    u[b * NN + r0 + threadIdx.x] = __logf(s);     // LSE (no max needed: x<=0)
  }
}

// ----------------------------------------------------------------- col pass
// One thread per column: fully coalesced streaming reads of C (L2-resident),
// 4 independent accumulators to keep the exp pipe full. Deterministic order.
__global__ __launch_bounds__(256) void sink_col(const float* __restrict__ C,
                                                const float* __restrict__ eps,
                                                const float* __restrict__ u,
                                                float* __restrict__ v) {
  const int b = blockIdx.y;
  const int j = blockIdx.x * 256 + threadIdx.x;
  const float inv_eps = 1.0f / eps[0];
  const float* Cb = C + (size_t)b * NN * NN + j;
  const float* ub = u + b * NN;
  float s0 = 0.f, s1 = 0.f, s2 = 0.f, s3 = 0.f;
#pragma unroll 4
  for (int i = 0; i < NN; i += 4) {
    s0 += __expf(fmaf(Cb[(size_t)(i + 0) * NN], -inv_eps, -ub[i + 0]));
    s1 += __expf(fmaf(Cb[(size_t)(i + 1) * NN], -inv_eps, -ub[i + 1]));
    s2 += __expf(fmaf(Cb[(size_t)(i + 2) * NN], -inv_eps, -ub[i + 2]));
    s3 += __expf(fmaf(Cb[(size_t)(i + 3) * NN], -inv_eps, -ub[i + 3]));
  }
  v[b * NN + j] = __logf((s0 + s1) + (s2 + s3));
}

// ----------------------------------------------------------------- output
__global__ __launch_bounds__(256) void sink_out(const float* __restrict__ C,
                                                const float* __restrict__ eps,
                                                const float* __restrict__ u,
                                                const float* __restrict__ v,
                                                float* __restrict__ out) {
  const size_t idx4 = ((size_t)blockIdx.x * 256 + threadIdx.x) * 4;
  const int b = (int)(idx4 >> 20);
  const int r = (int)(idx4 >> 10) & (NN - 1);
  const int j = (int)idx4 & (NN - 1);
  const float inv_eps = 1.0f / eps[0];
  const float ui = u[b * NN + r];
  const float4 c4 = *(const float4*)(C + idx4);
  const float4 v4 = *(const float4*)(v + b * NN + j);
  float4 o;
  o.x = __expf(fmaf(c4.x, -inv_eps, -ui - v4.x));
  o.y = __expf(fmaf(c4.y, -inv_eps, -ui - v4.y));
  o.z = __expf(fmaf(c4.z, -inv_eps, -ui - v4.z));
  o.w = __expf(fmaf(c4.w, -inv_eps, -ui - v4.w));
  *(float4*)(out + idx4) = o;
}

// ---------------------------------------------------------------------------
extern "C" void kernel_launch(void* const* d_in, const int* in_sizes, int n_in,
                              void* d_out, int out_size, void* d_ws, size_t ws_size,
                              hipStream_t stream) {
  const float* C   = (const float*)d_in[0];   // (64,1024,1024) f32
  const float* eps = (const float*)d_in[1];   // (1,) f32
  float* u = (float*)d_ws;                    // 64*1024 f32
  float* v = u + NB * NN;                     // 64*1024 f32

  sink_init<<<dim3(NB * NN / 256), dim3(256), 0, stream>>>(v);
  for (int t = 0; t < 20; ++t) {
    sink_row<<<dim3(NN / 16, NB), dim3(256), 0, stream>>>(C, eps, v, u);
    sink_col<<<dim3(NN / 256, NB), dim3(256), 0, stream>>>(C, eps, u, v);
  }
  sink_out<<<dim3((NB * NN * NN) / (256 * 4)), dim3(256), 0, stream>>>(
      C, eps, u, v, (float*)d_out);
}